// RIDE_42013370090071
// MI455X (gfx1250) — compile-verified
//
#include <hip/hip_runtime.h>

typedef float v2f __attribute__((ext_vector_type(2)));
typedef float v8f __attribute__((ext_vector_type(8)));

// ---------------- compile-time constants (temperature per class) -------------
struct RideConsts { float invT[7]; float scale; };

constexpr RideConsts make_consts() {
    double cls[7] = {705.0, 717.0, 281.0, 4772.0, 1982.0, 1290.0, 2524.0};
    double s = 0.0;
    for (int i = 0; i < 7; ++i) s += cls[i];
    double w[7] = {};
    double wmax = 0.0;
    for (int i = 0; i < 7; ++i) {
        w[i] = 7.0 * (cls[i] / s) * 0.015 + (1.0 - 0.015);
        if (w[i] > wmax) wmax = w[i];
    }
    RideConsts rc = {};
    double mT2 = 0.0;
    for (int i = 0; i < 7; ++i) {
        double T  = 1.5 * w[i] / wmax;
        float  Tf = (float)T;                  // reference casts T to f32
        rc.invT[i] = (float)(1.0 / (double)Tf);
        mT2 += (double)Tf * (double)Tf;
    }
    mT2 /= 7.0;                                 // mean over classes of T^2
    // loss = (sum over rows of [H - q.z_j + lse_j summed over j]) * mT2/(3*7*N)
    rc.scale = (float)(mT2 / (3.0 * 7.0 * 2097152.0));
    return rc;
}

#define N_ROWS   2097152
#define NBLK     1024
#define NTHR     256
#define ROWS_PER ((N_ROWS) / ((NBLK) * (NTHR)))   // == 8, exact

// ---------------- wave32 sum via V_WMMA_F32_16X16X4_F32 ----------------------
// A-VGPR0 = per-lane value, A-VGPR1 = 0, B = ones  =>  D[m,*] = v[m] + v[m+16].
// Sum of D's 8 VGPRs in lanes 0-15 covers m=0..7, lanes 16-31 cover m=8..15;
// one shfl_xor(16) add yields the full 32-lane sum (fp32 exact path).
__device__ __forceinline__ float wave_sum_wmma(float v) {
    v2f A; A.x = v;    A.y = 0.0f;
    v2f B; B.x = 1.0f; B.y = 1.0f;
    v8f C = {0.f, 0.f, 0.f, 0.f, 0.f, 0.f, 0.f, 0.f};
    v8f D = __builtin_amdgcn_wmma_f32_16x16x4_f32(false, A, false, B,
                                                  (short)0, C, false, false);
    float s = D[0] + D[1] + D[2] + D[3] + D[4] + D[5] + D[6] + D[7];
    s += __shfl_xor(s, 16, 32);
    return s;
}

// ---------------- per-head KL contribution: H - q.z + lse --------------------
// Inputs are streamed exactly once (235 MB > 192 MB L2): use non-temporal
// loads so the lines are not retained in the cache hierarchy.
__device__ __forceinline__ float head_term(const float* __restrict__ p, int base,
                                           const float* q, const float* invT,
                                           float H) {
    float z[7];
    #pragma unroll
    for (int c = 0; c < 7; ++c)
        z[c] = __builtin_nontemporal_load(p + base + c) * invT[c];
    float m = z[0];
    #pragma unroll
    for (int c = 1; c < 7; ++c) m = fmaxf(m, z[c]);
    float S = 0.0f, dot = 0.0f;
    #pragma unroll
    for (int c = 0; c < 7; ++c) { S += __expf(z[c] - m); dot += q[c] * z[c]; }
    float lse = m + __logf(S);
    return H - dot + lse;
}

// ---------------- kernel 1: stream rows, per-block partial sums --------------
__global__ __launch_bounds__(NTHR) void ride_partial_kernel(
    const float* __restrict__ o1, const float* __restrict__ o2,
    const float* __restrict__ o3, const float* __restrict__ o4,
    float* __restrict__ partial)
{
    constexpr RideConsts cc = make_consts();
    const int tid    = blockIdx.x * NTHR + threadIdx.x;
    const int stride = NBLK * NTHR;

    float acc = 0.0f;
    #pragma unroll
    for (int k = 0; k < ROWS_PER; ++k) {
        const int row  = tid + k * stride;      // < 2^21, int is safe
        const int base = row * 7;               // < 2^25

        // Teacher: softmax/log-softmax of o1/T (non-temporal streaming reads)
        float z0[7];
        #pragma unroll
        for (int c = 0; c < 7; ++c)
            z0[c] = __builtin_nontemporal_load(o1 + base + c) * cc.invT[c];
        float m0 = z0[0];
        #pragma unroll
        for (int c = 1; c < 7; ++c) m0 = fmaxf(m0, z0[c]);
        float e[7], S = 0.0f;
        #pragma unroll
        for (int c = 0; c < 7; ++c) { e[c] = __expf(z0[c] - m0); S += e[c]; }
        const float rS = 1.0f / S;
        const float lS = __logf(S);
        float q[7], H = 0.0f;                   // H = sum q*logq
        #pragma unroll
        for (int c = 0; c < 7; ++c) {
            q[c] = e[c] * rS;
            H += q[c] * ((z0[c] - m0) - lS);
        }

        float cost = head_term(o2, base, q, cc.invT, H);
        cost      += head_term(o3, base, q, cc.invT, H);
        cost      += head_term(o4, base, q, cc.invT, H);
        acc += cost;
    }

    // Wave reduction on the matrix pipe (EXEC is all-ones: no divergence above)
    const float wave_total = wave_sum_wmma(acc);

    __shared__ float wsum[NTHR / 32];
    const int lane = threadIdx.x & 31;
    const int wid  = threadIdx.x >> 5;
    if (lane == 0) wsum[wid] = wave_total;
    __syncthreads();
    if (threadIdx.x == 0) {
        float t = 0.0f;
        #pragma unroll
        for (int i = 0; i < NTHR / 32; ++i) t += wsum[i];
        partial[blockIdx.x] = t;       // RT path: stays in L2 for kernel 2
    }
}

// ---------------- kernel 2: WMMA-accumulated final reduction -----------------
// One wave; each WMMA folds 64 partials (A = 2 VGPRs x 32 lanes) into D with
// C-accumulation. Fixed order -> deterministic.
__global__ __launch_bounds__(32) void ride_final_kernel(
    const float* __restrict__ partial, float* __restrict__ out)
{
    constexpr RideConsts cc = make_consts();
    const int lane = threadIdx.x;

    v2f B; B.x = 1.0f; B.y = 1.0f;
    v8f D = {0.f, 0.f, 0.f, 0.f, 0.f, 0.f, 0.f, 0.f};
    for (int i = 0; i < NBLK; i += 64) {
        v2f A; A.x = partial[i + lane]; A.y = partial[i + 32 + lane];
        D = __builtin_amdgcn_wmma_f32_16x16x4_f32(false, A, false, B,
                                                  (short)0, D, false, false);
    }
    float s = D[0] + D[1] + D[2] + D[3] + D[4] + D[5] + D[6] + D[7];
    s += __shfl_xor(s, 16, 32);
    if (lane == 0) out[0] = s * cc.scale;
}

// ---------------- launch -----------------------------------------------------
extern "C" void kernel_launch(void* const* d_in, const int* in_sizes, int n_in,
                              void* d_out, int out_size, void* d_ws, size_t ws_size,
                              hipStream_t stream) {
    const float* o1 = (const float*)d_in[0];
    const float* o2 = (const float*)d_in[1];
    const float* o3 = (const float*)d_in[2];
    const float* o4 = (const float*)d_in[3];
    // d_in[4] (targets) is unused by the reference loss math.
    float* partial = (float*)d_ws;            // NBLK floats = 4 KB of scratch
    float* out     = (float*)d_out;

    ride_partial_kernel<<<NBLK, NTHR, 0, stream>>>(o1, o2, o3, o4, partial);
    ride_final_kernel<<<1, 32, 0, stream>>>(partial, out);
}